// ROIPool_3d_90469191123539
// MI455X (gfx1250) — compile-verified
//
#include <hip/hip_runtime.h>
#include <cstdint>

// ROIPool3D (fixed 8x16x16 crop -> 4x4x4 adaptive max) for MI455X / gfx1250.
// Strategy: one TDM tensor_load_to_lds per block DMAs a 4D tile
// (W=16, H=16, D=8, C=4) of the feature map into LDS (with padding to a
// 17-dword row pitch for bank-conflict-free pooling reads), then 256 threads
// each max-reduce one 2x4x4 window and store one output float (coalesced).
// Problem is pure bandwidth (~67MB in, 2MB out -> ~3us at 23.3 TB/s); the TDM
// path moves the strided crop rows without burning VALU/VGPRs on addressing.

#define C_TOT   64
#define D_TOT   64
#define H_TOT   128
#define W_TOT   128
#define HW      (H_TOT * W_TOT)          // 16384
#define DHW     (D_TOT * HW)             // 1048576
#define CG      4                        // channels per block
#define ROW_PITCH 17                     // 16 dwords + 1 pad dword
#define CH_PITCH  (128 * ROW_PITCH)      // 8*16 rows per channel = 2176 dwords

typedef int v4i __attribute__((ext_vector_type(4)));
typedef int v8i __attribute__((ext_vector_type(8)));

__global__ __launch_bounds__(256)
void roipool3d_tdm_kernel(const float* __restrict__ in,
                          const int* __restrict__ rois,
                          float* __restrict__ out)
{
    __shared__ float lds[CG * CH_PITCH];   // 34,816 bytes

    const int cg  = blockIdx.x;            // channel group 0..15
    const int roi = blockIdx.y;            // 0..127
    const int t   = threadIdx.x;

    // ---- Wave 0: build Tensor DMA descriptor (D#) and issue TDM ----
    if (t < 32) {
        const int* rp = rois + roi * 7;    // [b, z1, y1, x1, z2, y2, x2]
        const int z1 = __builtin_amdgcn_readfirstlane(rp[1]);
        const int y1 = __builtin_amdgcn_readfirstlane(rp[2]);
        const int x1 = __builtin_amdgcn_readfirstlane(rp[3]);

        const unsigned long long gaddr =
            (unsigned long long)(uintptr_t)in +
            4ull * ((unsigned long long)(cg * CG) * DHW +
                    (unsigned long long)z1 * HW +
                    (unsigned long long)y1 * W_TOT +
                    (unsigned long long)x1);
        const unsigned lds_base = (unsigned)(uintptr_t)&lds[0]; // low 32b = LDS byte offset

        // D# group 0: count=1 | lds_addr | global_addr[56:0] | type=2
        v4i g0;
        g0.x = 1;
        g0.y = (int)__builtin_amdgcn_readfirstlane((int)lds_base);
        g0.z = (int)__builtin_amdgcn_readfirstlane((int)(unsigned)gaddr);
        g0.w = (int)__builtin_amdgcn_readfirstlane(
                   (int)(((unsigned)(gaddr >> 32) & 0x01FFFFFFu) | 0x80000000u));

        // D# group 1:
        //  [17:16] data_size=2 (4B), [20] pad_enable=1,
        //  [24:22] pad_interval=3 (every 16 dwords), [31:25] pad_amount=0 (1 dword)
        //  tensor_dim0=16, tensor_dim1=16, tile_dim0=16, tile_dim1=16, tile_dim2=8
        //  tensor_dim0_stride=W(128), tensor_dim1_stride=HW(16384)
        v8i g1;
        g1[0] = (2 << 16) | (1 << 20) | (3 << 22);
        g1[1] = (16 << 16);                 // tensor_dim0[15:0] in bits 63:48
        g1[2] = (16 << 16);                 // tensor_dim0 hi=0; tensor_dim1[15:0]
        g1[3] = (16 << 16);                 // tensor_dim1 hi=0; tile_dim0=16
        g1[4] = 16 | (8 << 16);             // tile_dim1=16, tile_dim2=8
        g1[5] = W_TOT;                      // tensor_dim0_stride lo32
        g1[6] = (HW & 0xFFFF) << 16;        // stride0 hi=0; tensor_dim1_stride[15:0]
        g1[7] = HW >> 16;                   // tensor_dim1_stride[47:16] (=0)

        // D# group 2: tensor_dim2=8, tensor_dim3=CG, tensor_dim2_stride=DHW, tile_dim3=CG
        v4i g2;
        g2.x = 8;
        g2.y = CG;
        g2.z = DHW;                         // stride lo32
        g2.w = (CG << 16);                  // stride hi=0; tile_dim3

        // D# group 3: tensor_dim3_stride=DHW, dim4 unused
        v4i g3;
        g3.x = DHW;
        g3.y = 0;
        g3.z = 0;
        g3.w = 0;

        asm volatile("tensor_load_to_lds %0, %1, %2, %3"
                     :
                     : "s"(g0), "s"(g1), "s"(g2), "s"(g3)
                     : "memory");
        __builtin_amdgcn_s_wait_tensorcnt(0);
    }
    __syncthreads();

    // ---- Pooling: one output per thread ----
    const int cl = t >> 6;                  // local channel 0..3
    const int s  = t & 63;                  // od*16 + oh*4 + ow
    const int od = s >> 4;
    const int oh = (s >> 2) & 3;
    const int ow = s & 3;

    const float* ch = lds + cl * CH_PITCH;
    const int r0 = (od * 2) * 16 + oh * 4;  // base row index (d*16 + h)

    float m = -3.402823466e38f;
#pragma unroll
    for (int dd = 0; dd < 2; ++dd) {
#pragma unroll
        for (int hh = 0; hh < 4; ++hh) {
            const float* p = ch + (r0 + dd * 16 + hh) * ROW_PITCH + ow * 4;
            m = fmaxf(m, p[0]);
            m = fmaxf(m, p[1]);
            m = fmaxf(m, p[2]);
            m = fmaxf(m, p[3]);
        }
    }

    // out[roi][c][od][oh][ow], c = cg*4 + cl
    out[((roi * C_TOT + cg * CG + cl) << 6) + s] = m;
}

extern "C" void kernel_launch(void* const* d_in, const int* in_sizes, int n_in,
                              void* d_out, int out_size, void* d_ws, size_t ws_size,
                              hipStream_t stream) {
    const float* in   = (const float*)d_in[0];
    const int*   rois = (const int*)d_in[1];
    float*       out  = (float*)d_out;

    dim3 grid(C_TOT / CG, 128);   // (16 channel-groups, 128 ROIs)
    roipool3d_tdm_kernel<<<grid, 256, 0, stream>>>(in, rois, out);
}